// my_TGCN_80504866996870
// MI455X (gfx1250) — compile-verified
//
#include <hip/hip_runtime.h>
#include <math.h>

typedef float v2f __attribute__((ext_vector_type(2)));
typedef float v8f __attribute__((ext_vector_type(8)));

#define NB 8     // graphs
#define F  64    // IN == OUT == 64

// ---------------- init: deg = 1 (self loop), S = 0, cnt = 0 ----------------
__global__ void k_init(float* __restrict__ deg, float* __restrict__ S,
                       float* __restrict__ cnt, int n) {
  int i = blockIdx.x * blockDim.x + threadIdx.x;
  if (i < n)      deg[i] = 1.0f;
  if (i < NB * F) S[i]   = 0.0f;
  if (i < NB)     cnt[i] = 0.0f;
}

// ---------------- degree accumulation over edges ----------------
__global__ void k_deg(const int* __restrict__ col, const float* __restrict__ ew,
                      float* __restrict__ deg, int E) {
  int stride = gridDim.x * blockDim.x;
  for (int e = blockIdx.x * blockDim.x + threadIdx.x; e < E; e += stride)
    atomicAdd(&deg[col[e]], ew[e]);
}

// ------- per-node: deg -> dinv (in place), self-loop X[i]/deg into S, counts -------
__global__ void k_node(const float* __restrict__ X, const int* __restrict__ batch,
                       float* __restrict__ deg, float* __restrict__ Sg,
                       float* __restrict__ cntg, int n) {
  __shared__ float Sl[NB * F];
  __shared__ float cl[NB];
  int t = threadIdx.x;
  for (int i = t; i < NB * F; i += blockDim.x) Sl[i] = 0.0f;
  if (t < NB) cl[t] = 0.0f;
  __syncthreads();
  int g = t >> 4, j = t & 15;               // 16 lanes cooperate on one node
  int gpb = blockDim.x >> 4;
  int stride = gridDim.x * gpb;
  for (int i = blockIdx.x * gpb + g; i < n; i += stride) {
    float d = deg[i];                        // >= 1 always (self loop)
    float dinv = 1.0f / sqrtf(d);
    int b = batch[i];
    if (j == 0) { deg[i] = dinv; atomicAdd(&cl[b], 1.0f); }
    float sc = 1.0f / d;                     // dinv^2
    float4 x = ((const float4*)(X + (size_t)i * F))[j];
    atomicAdd(&Sl[b * F + j * 4 + 0], sc * x.x);
    atomicAdd(&Sl[b * F + j * 4 + 1], sc * x.y);
    atomicAdd(&Sl[b * F + j * 4 + 2], sc * x.z);
    atomicAdd(&Sl[b * F + j * 4 + 3], sc * x.w);
  }
  __syncthreads();
  for (int i = t; i < NB * F; i += blockDim.x) atomicAdd(&Sg[i], Sl[i]);
  if (t < NB) atomicAdd(&cntg[t], cl[t]);
}

// ---------------- edge pass: S[batch[col]] += norm * X[row] ----------------
__global__ void k_edge(const float* __restrict__ X, const int* __restrict__ row,
                       const int* __restrict__ col, const float* __restrict__ ew,
                       const float* __restrict__ dinv, const int* __restrict__ batch,
                       float* __restrict__ Sg, int E) {
  __shared__ float Sl[NB * F];
  int t = threadIdx.x;
  for (int i = t; i < NB * F; i += blockDim.x) Sl[i] = 0.0f;
  __syncthreads();
  int g = t >> 4, j = t & 15;               // 16 lanes per edge -> 256B coalesced gather
  int gpb = blockDim.x >> 4;
  int stride = gridDim.x * gpb;
  for (int e = blockIdx.x * gpb + g; e < E; e += stride) {
    int r = row[e], c = col[e];
    float nrm = dinv[r] * ew[e] * dinv[c];
    int b = batch[c];
    float4 x = ((const float4*)(X + (size_t)r * F))[j];
    atomicAdd(&Sl[b * F + j * 4 + 0], nrm * x.x);
    atomicAdd(&Sl[b * F + j * 4 + 1], nrm * x.y);
    atomicAdd(&Sl[b * F + j * 4 + 2], nrm * x.z);
    atomicAdd(&Sl[b * F + j * 4 + 3], nrm * x.w);
  }
  __syncthreads();
  for (int i = t; i < NB * F; i += blockDim.x) atomicAdd(&Sg[i], Sl[i]);
}

// ---------------- finalize: fp32 WMMA GEMM chain + GRU blend ----------------
__device__ __forceinline__ v8f wmma4(v2f a, v2f b, v8f c) {
  return __builtin_amdgcn_wmma_f32_16x16x4_f32(
      /*neg_a=*/false, a, /*neg_b=*/false, b, /*c_mod=*/(short)0, c,
      /*reuse_a=*/false, /*reuse_b=*/false);
}

__global__ void __launch_bounds__(128) k_final(
    const float* __restrict__ Sg, const float* __restrict__ cnt,
    const float* __restrict__ H,
    const float* __restrict__ Wz, const float* __restrict__ bz,
    const float* __restrict__ Lz, const float* __restrict__ lbz,
    const float* __restrict__ Wh, const float* __restrict__ bh,
    const float* __restrict__ Lh, const float* __restrict__ lbh,
    float* __restrict__ out) {
  __shared__ float poolL[16 * F];   // pool, M padded 8 -> 16 with zeros
  __shared__ float Hs[16 * F];      // H, padded
  __shared__ float Tg[16 * F];      // per-gate intermediate T = pool@W + b
  __shared__ float Zs[16 * F];
  __shared__ float Hts[16 * F];
  int t = threadIdx.x;
  for (int i = t; i < 16 * F; i += 128) {
    int m = i >> 6;
    float pv = 0.0f, hv = 0.0f;
    if (m < NB) {
      float c = cnt[m]; c = c > 1.0f ? c : 1.0f;
      pv = Sg[i] / c;
      hv = H[i];
    }
    poolL[i] = pv;
    Hs[i] = hv;
  }
  __syncthreads();

  int lane  = t & 31;
  int wave  = t >> 5;          // 4 waves -> 4 N-tiles of 16
  int m16   = lane & 15;       // A row / B,C,D column (within tile)
  int khalf = lane >> 4;       // K sub-block selector
  int n     = m16 + wave * 16; // global column 0..63

  auto run_gate = [&](const float* W, const float* bias, const float* L,
                      const float* lb, float* dst, bool use_tanh) {
    // stage 1: T = pool @ W + bias
    v8f c = {0.f, 0.f, 0.f, 0.f, 0.f, 0.f, 0.f, 0.f};
#pragma unroll
    for (int kk = 0; kk < 16; ++kk) {
      int kb = 4 * kk + khalf * 2;
      v2f a, b;
      a.x = poolL[m16 * F + kb];
      a.y = poolL[m16 * F + kb + 1];
      b.x = W[kb * F + n];
      b.y = W[(kb + 1) * F + n];
      c = wmma4(a, b, c);
    }
    float bb = bias[n];
#pragma unroll
    for (int v = 0; v < 8; ++v)
      Tg[(v + khalf * 8) * F + n] = c[v] + bb;   // C/D layout: row = v + (lane/16)*8
    __syncthreads();

    // stage 2: U = T @ L[0:64] + H @ L[64:128] + lb ; activation
    v8f u = {0.f, 0.f, 0.f, 0.f, 0.f, 0.f, 0.f, 0.f};
#pragma unroll
    for (int kk = 0; kk < 16; ++kk) {
      int kb = 4 * kk + khalf * 2;
      v2f a, b;
      a.x = Tg[m16 * F + kb];
      a.y = Tg[m16 * F + kb + 1];
      b.x = L[kb * F + n];
      b.y = L[(kb + 1) * F + n];
      u = wmma4(a, b, u);
    }
#pragma unroll
    for (int kk = 0; kk < 16; ++kk) {
      int kb = 4 * kk + khalf * 2;
      v2f a, b;
      a.x = Hs[m16 * F + kb];
      a.y = Hs[m16 * F + kb + 1];
      b.x = L[(64 + kb) * F + n];
      b.y = L[(64 + kb + 1) * F + n];
      u = wmma4(a, b, u);
    }
    float lbb = lb[n];
#pragma unroll
    for (int v = 0; v < 8; ++v) {
      float x = u[v] + lbb;
      float r = use_tanh ? tanhf(x) : 1.0f / (1.0f + expf(-x));
      dst[(v + khalf * 8) * F + n] = r;
    }
    __syncthreads();
  };

  run_gate(Wz, bz, Lz, lbz, Zs, false);   // Z = sigmoid(...)
  run_gate(Wh, bh, Lh, lbh, Hts, true);   // H_tilde = tanh(...)
  // (r gate intentionally skipped: it does not affect the output)

  for (int i = t; i < NB * F; i += 128) {
    float z = Zs[i];
    out[i] = z * Hs[i] + (1.0f - z) * Hts[i];
  }
}

extern "C" void kernel_launch(void* const* d_in, const int* in_sizes, int n_in,
                              void* d_out, int out_size, void* d_ws, size_t ws_size,
                              hipStream_t stream) {
  (void)n_in; (void)out_size; (void)ws_size;
  const float* X    = (const float*)d_in[0];
  const int*   ei   = (const int*)d_in[1];
  const int*   bat  = (const int*)d_in[2];
  const float* ew   = (const float*)d_in[3];
  const float* H    = (const float*)d_in[4];
  const float* Wz   = (const float*)d_in[5];
  const float* bz   = (const float*)d_in[6];
  const float* Lz   = (const float*)d_in[7];
  const float* lbz  = (const float*)d_in[8];
  // d_in[9..12] = r-gate params: unused (dead in reference output)
  const float* Wh   = (const float*)d_in[13];
  const float* bh   = (const float*)d_in[14];
  const float* Lh   = (const float*)d_in[15];
  const float* lbh  = (const float*)d_in[16];

  int N = in_sizes[0] / F;
  int E = in_sizes[3];
  const int* row = ei;          // edge_index[0]
  const int* col = ei + E;      // edge_index[1]

  float* deg = (float*)d_ws;    // [N]  (becomes dinv in place)
  float* S   = deg + N;         // [8*64]
  float* cnt = S + NB * F;      // [8]

  k_init<<<(N + 255) / 256, 256, 0, stream>>>(deg, S, cnt, N);
  k_deg <<<1024, 256, 0, stream>>>(col, ew, deg, E);
  k_node<<<512, 256, 0, stream>>>(X, bat, deg, S, cnt, N);
  k_edge<<<1024, 256, 0, stream>>>(X, row, col, ew, deg, bat, S, E);
  k_final<<<1, 128, 0, stream>>>(S, cnt, H, Wz, bz, Lz, lbz, Wh, bh, Lh, lbh,
                                 (float*)d_out);
}